// SGE_8220567405268
// MI455X (gfx1250) — compile-verified
//
#include <hip/hip_runtime.h>

typedef __attribute__((ext_vector_type(2))) float v2f;
typedef __attribute__((ext_vector_type(8))) float v8f;

#define FEAT 128
#define EMB  256
#define NPTS 32
#define SDIM 8
#define NEPS 9

__constant__ float EPS_SCHED[NEPS] = {
    10.0f, 5.0f, 2.5f, 1.25f, 0.625f, 0.3125f, 0.15625f, 0.078125f, 0.05f
};

// ---------------------------------------------------------------------------
// 1) scatter: agg[dst] += x[src], deg[dst] += 1   (mean-aggregation GCN)
// ---------------------------------------------------------------------------
__global__ __launch_bounds__(256)
void sge_scatter(const float* __restrict__ x, const long long* __restrict__ ei,
                 float* __restrict__ agg, float* __restrict__ deg, int E)
{
    long long t = (long long)blockIdx.x * blockDim.x + threadIdx.x;
    long long e = t >> 5;            // 32 groups of 4 features per edge
    int g = (int)(t & 31);
    if (e >= E) return;
    long long src = ei[e];
    long long dst = ei[(long long)E + e];
    // prefetch a future edge's source row (random rows, L2-resident x)
    if (g == 0 && e + 64 < E) {
        long long psrc = ei[e + 64];
        __builtin_prefetch(x + psrc * FEAT, 0, 1);
    }
    const float4 v = *(const float4*)(x + src * FEAT + g * 4);
    float* a = agg + dst * FEAT + g * 4;
    atomicAdd(a + 0, v.x);
    atomicAdd(a + 1, v.y);
    atomicAdd(a + 2, v.z);
    atomicAdd(a + 3, v.w);
    if (g == 0) atomicAdd(deg + dst, 1.0f);
}

// ---------------------------------------------------------------------------
// 2) z = (agg / max(deg,1)) @ W   via V_WMMA_F32_16X16X4_F32
//    one wave per 16x16 output tile; M=100000 -> 6250 tiles, N=256 -> 16 tiles
// ---------------------------------------------------------------------------
__global__ __launch_bounds__(256)
void sge_gemm_wmma(const float* __restrict__ agg, const float* __restrict__ deg,
                   const float* __restrict__ W, float* __restrict__ z, int Mtiles)
{
    int wid  = (blockIdx.x * blockDim.x + threadIdx.x) >> 5;
    int lane = threadIdx.x & 31;
    int tm = wid >> 4;               // EMB/16 == 16 tiles along N
    int tn = wid & 15;
    if (tm >= Mtiles) return;        // wave-uniform guard: EXEC stays all-ones
    int hh = lane >> 4;              // half-wave select (K pair)
    int lm = lane & 15;
    int row = tm * 16 + lm;          // A row for this lane (both halves)
    int col = tn * 16 + lm;          // B/D column for this lane
    float invdeg = 1.0f / fmaxf(deg[row], 1.0f);
    // A fragment: lane loads exactly its K pair -> single b64 load, no selects
    const float* A = agg + (long long)row * FEAT + 2 * hh;
    const float* Bp = W + 2 * hh * EMB + col;

    v8f c = {};
    #pragma unroll 4
    for (int k = 0; k < FEAT; k += 4) {
        float2 av = *(const float2*)(A + k);
        v2f a, b;
        a[0] = av.x * invdeg;
        a[1] = av.y * invdeg;
        b[0] = Bp[(long long)k * EMB];
        b[1] = Bp[(long long)(k + 1) * EMB];
        c = __builtin_amdgcn_wmma_f32_16x16x4_f32(
                /*neg_a=*/false, a, /*neg_b=*/false, b,
                /*c_mod=*/(short)0, c, /*reuse_a=*/false, /*reuse_b=*/false);
    }
    // D 16x16 f32 layout: VGPR r -> M = r + 8*half, N = lane%16
    #pragma unroll
    for (int r = 0; r < 8; ++r)
        z[(long long)(tm * 16 + r + 8 * hh) * EMB + col] = c[r];
}

// ---------------------------------------------------------------------------
// 3) debiased Sinkhorn divergence, one wave per edge pair.
//    Cost matrices via WMMA Gram tiles: C_ij = sqrt(max(|x|^2+|y|^2-2x.y,0)+1e-8)
// ---------------------------------------------------------------------------
__device__ __forceinline__ v8f sge_gram_tile(const float (*Ls)[SDIM],
                                             const float (*Rs)[SDIM],
                                             int Mt, int Nt, int hh, int lm)
{
    v8f c = {};
    #pragma unroll
    for (int kk = 0; kk < SDIM; kk += 4) {
        v2f a, b;
        // A 16x4 f32 layout: lanes 0-15 K=kk,kk+1 ; lanes 16-31 K=kk+2,kk+3
        a[0] = Ls[Mt * 16 + lm][kk + 2 * hh];
        a[1] = Ls[Mt * 16 + lm][kk + 2 * hh + 1];
        // B 4x16 of R^T: B[k][n] = R[n][k] -> same per-lane pattern on R rows
        b[0] = Rs[Nt * 16 + lm][kk + 2 * hh];
        b[1] = Rs[Nt * 16 + lm][kk + 2 * hh + 1];
        c = __builtin_amdgcn_wmma_f32_16x16x4_f32(
                false, a, false, b, (short)0, c, false, false);
    }
    return c;
}

__device__ __forceinline__ void sge_cost_matrix(const float (*Ls)[SDIM],
                                                const float (*Rs)[SDIM],
                                                const float* nL, const float* nR,
                                                float* C, int hh, int lm)
{
    #pragma unroll
    for (int Mt = 0; Mt < 2; ++Mt) {
        #pragma unroll
        for (int Nt = 0; Nt < 2; ++Nt) {
            v8f c = sge_gram_tile(Ls, Rs, Mt, Nt, hh, lm);
            int j = Nt * 16 + lm;
            float nr = nR[j];
            #pragma unroll
            for (int r = 0; r < 8; ++r) {
                int i = Mt * 16 + r + 8 * hh;
                float sq = nL[i] + nr - 2.0f * c[r];
                C[i * NPTS + j] = sqrtf(fmaxf(sq, 0.0f) + 1e-8f);
            }
        }
    }
}

__device__ __forceinline__ float sge_softmin(float eps, float inv_eps,
                                             const float* __restrict__ C, int stride,
                                             const float* __restrict__ H)
{
    const float LOGN = 3.4657359028f;    // log(32)
    float m = -1e30f;
    for (int j = 0; j < NPTS; ++j) {
        float v = (H[j] - C[j * stride]) * inv_eps;
        m = fmaxf(m, v);
    }
    float s = 0.0f;
    for (int j = 0; j < NPTS; ++j) {
        float v = (H[j] - C[j * stride]) * inv_eps;
        s += __expf(v - m);
    }
    return -eps * (m + __logf(s) - LOGN);
}

__global__ __launch_bounds__(32)
void sge_sinkhorn(const float* __restrict__ z, const long long* __restrict__ edges,
                  int P, float* __restrict__ energy)
{
    __shared__ float Cxy[NPTS * NPTS], Cxx[NPTS * NPTS], Cyy[NPTS * NPTS];
    __shared__ float Xs[NPTS][SDIM], Ys[NPTS][SDIM];
    __shared__ float nX[NPTS], nY[NPTS];
    __shared__ float Fab[NPTS], Gba[NPTS], Faa[NPTS], Gbb[NPTS];

    int e = blockIdx.x;
    int lane = threadIdx.x;
    int hh = lane >> 4;
    int lm = lane & 15;
    long long nxi = edges[e];
    long long nyi = edges[(long long)P + e];
    const float* px = z + nxi * EMB + lane * SDIM;
    const float* py = z + nyi * EMB + lane * SDIM;
    float sx = 0.0f, sy = 0.0f;
    #pragma unroll
    for (int d = 0; d < SDIM; ++d) {
        float xv = px[d], yv = py[d];
        Xs[lane][d] = xv; Ys[lane][d] = yv;
        sx += xv * xv;    sy += yv * yv;
    }
    nX[lane] = sx;
    nY[lane] = sy;
    __syncthreads();

    // cost matrices on the matrix pipe (24 WMMAs); EXEC is all-ones here
    sge_cost_matrix(Xs, Ys, nX, nY, Cxy, hh, lm);
    sge_cost_matrix(Xs, Xs, nX, nX, Cxx, hh, lm);
    sge_cost_matrix(Ys, Ys, nY, nY, Cyy, hh, lm);
    __syncthreads();

    float f_ab = 0.f, g_ba = 0.f, f_aa = 0.f, g_bb = 0.f;
    for (int t = 0; t < NEPS; ++t) {
        float eps = EPS_SCHED[t];
        float inv_eps = 1.0f / eps;
        Fab[lane] = f_ab; Gba[lane] = g_ba; Faa[lane] = f_aa; Gbb[lane] = g_bb;
        __syncthreads();
        float ft  = sge_softmin(eps, inv_eps, &Cxy[lane * NPTS], 1,    Gba); // rows of C_xy
        float gt  = sge_softmin(eps, inv_eps, &Cxy[lane],        NPTS, Fab); // C_yx = C_xy^T
        float fat = sge_softmin(eps, inv_eps, &Cxx[lane * NPTS], 1,    Faa);
        float gbt = sge_softmin(eps, inv_eps, &Cyy[lane * NPTS], 1,    Gbb);
        __syncthreads();
        f_ab = 0.5f * (f_ab + ft);
        g_ba = 0.5f * (g_ba + gt);
        f_aa = 0.5f * (f_aa + fat);
        g_bb = 0.5f * (g_bb + gbt);
    }

    float val = (f_ab - f_aa) + (g_ba - g_bb);
    #pragma unroll
    for (int off = 16; off > 0; off >>= 1)
        val += __shfl_xor(val, off, 32);
    if (lane == 0) energy[e] = val * (1.0f / NPTS);
}

// ---------------------------------------------------------------------------
// 4) out = mean(pos^2) + mean(max(1-neg,0)^2)
// ---------------------------------------------------------------------------
__global__ __launch_bounds__(256)
void sge_reduce(const float* __restrict__ epos, const float* __restrict__ eneg,
                int P, float* __restrict__ out)
{
    __shared__ float sm[256];
    float acc = 0.0f;
    for (int i = threadIdx.x; i < P; i += 256) {
        float p = epos[i];
        float m = fmaxf(1.0f - eneg[i], 0.0f);
        acc += p * p + m * m;
    }
    sm[threadIdx.x] = acc;
    __syncthreads();
    for (int s = 128; s > 0; s >>= 1) {
        if (threadIdx.x < s) sm[threadIdx.x] += sm[threadIdx.x + s];
        __syncthreads();
    }
    if (threadIdx.x == 0) out[0] = sm[0] / (float)P;
}

extern "C" void kernel_launch(void* const* d_in, const int* in_sizes, int n_in,
                              void* d_out, int out_size, void* d_ws, size_t ws_size,
                              hipStream_t stream)
{
    const float*     x  = (const float*)d_in[0];
    const long long* ei = (const long long*)d_in[1];   // int64 per reference
    const long long* ep = (const long long*)d_in[2];
    const long long* en = (const long long*)d_in[3];
    const float*     W  = (const float*)d_in[4];

    const int N = in_sizes[0] / FEAT;        // 100000 nodes
    const int E = in_sizes[1] / 2;           // 1.6M message edges
    const int P = in_sizes[2] / 2;           // 30000 eval edges per set

    float* agg  = (float*)d_ws;                       // [N, FEAT]
    float* deg  = agg  + (size_t)N * FEAT;            // [N]
    float* z    = deg  + N;                           // [N, EMB]
    float* epos = z    + (size_t)N * EMB;             // [P]
    float* eneg = epos + P;                           // [P]

    // zero agg + deg (contiguous)
    hipMemsetAsync(d_ws, 0, ((size_t)N * FEAT + N) * sizeof(float), stream);

    // scatter-sum + degree
    {
        long long threads = (long long)E * 32;
        int blocks = (int)((threads + 255) / 256);
        sge_scatter<<<blocks, 256, 0, stream>>>(x, ei, agg, deg, E);
    }

    // z = (agg/deg) @ W with f32 WMMA; 16x16 tile per wave
    {
        int Mtiles = (N + 15) / 16;                  // 6250
        int waves  = Mtiles * (EMB / 16);            // 100000
        int blocks = (waves + 7) / 8;                // 8 waves / 256-thread block
        sge_gemm_wmma<<<blocks, 256, 0, stream>>>(agg, deg, W, z, Mtiles);
    }

    // Sinkhorn divergences for positive and negative edge sets
    sge_sinkhorn<<<P, 32, 0, stream>>>(z, ep, P, epos);
    sge_sinkhorn<<<P, 32, 0, stream>>>(z, en, P, eneg);

    // final scalar
    sge_reduce<<<1, 256, 0, stream>>>(epos, eneg, P, (float*)d_out);
}